// CompressKV_31155692765753
// MI455X (gfx1250) — compile-verified
//
#include <hip/hip_runtime.h>

typedef __attribute__((ext_vector_type(2))) float v2f;
typedef __attribute__((ext_vector_type(8))) float v8f;

#define KSZ    32
#define BATCH  2
#define NROWS  8192
#define DDIM   128
#define CCH    2048      // H*D = 16*128
#define MOUT   511       // (8192-32)/16 + 1

// One wave computes a 16(m) x 16(channel) f32 output tile as a banded matmul
// over the 272 unique x rows it touches, using V_WMMA_F32_16X16X4_F32.
__global__ __launch_bounds__(128) void compresskv_wmma_kernel(
    const float* __restrict__ x, const float* __restrict__ w,
    const float* __restrict__ pe, float* __restrict__ y)
{
    const int lane = threadIdx.x & 31;
    const int wave = threadIdx.x >> 5;
    const int ct   = blockIdx.x * 4 + wave;   // 0..127  channel tile
    const int mt   = blockIdx.y;              // 0..31   m tile
    const int b    = blockIdx.z;              // 0..1

    const int col  = lane & 15;               // B/D column, A row index
    const int half = lane >> 4;               // which half-wave
    const int koff = half * 2;                // K offset of this lane's 2 regs
    const int mrow = col;                     // A-matrix row this lane holds

    const int  c     = ct * 16 + col;         // flat channel 0..2047
    const long xbase = (long)b * NROWS * CCH;

    // Weights this lane needs for its A fragments (t = 4s + koff {, +1})
    float wA0[4], wA1[4], wB0[4], wB1[4];
#pragma unroll
    for (int s = 0; s < 4; ++s) {
        const int t = 4 * s + koff;
        wA0[s] = w[t];        wA1[s] = w[t + 1];
        wB0[s] = w[16 + t];   wB1[s] = w[16 + t + 1];
    }

    v8f acc = {0.f, 0.f, 0.f, 0.f, 0.f, 0.f, 0.f, 0.f};

    const int row0 = mt * 256;                // first x row of this m-tile
#pragma unroll 1
    for (int j = 0; j <= 16; ++j) {           // 17 chunks of 16 rows
        const int rbase = row0 + j * 16;

        // Prefetch next chunk's first row segment for this lane.
        if (j < 16) {
            int pr = rbase + 16 + koff;
            pr = pr < NROWS ? pr : NROWS - 1;
            __builtin_prefetch(&x[xbase + (long)pr * CCH + c], 0, 1);
        }

        // B fragments: 4 K-slices of the 16x16 (row x channel) chunk
        v2f bf[4];
#pragma unroll
        for (int s = 0; s < 4; ++s) {
            int r0 = rbase + 4 * s + koff;
            int r1 = r0 + 1;
            r0 = r0 < NROWS ? r0 : NROWS - 1;   // only last tile can exceed;
            r1 = r1 < NROWS ? r1 : NROWS - 1;   // clamped rows feed masked m
            bf[s].x = x[xbase + (long)r0 * CCH + c];
            bf[s].y = x[xbase + (long)r1 * CCH + c];
        }

        // A fragments: banded conv matrix. Chunk j hits row i=j with w[0:16]
        // and row i=j-1 with w[16:32]; everything else is zero.
#pragma unroll
        for (int s = 0; s < 4; ++s) {
            v2f af;
            af.x = (mrow == j) ? wA0[s] : ((mrow == j - 1) ? wB0[s] : 0.f);
            af.y = (mrow == j) ? wA1[s] : ((mrow == j - 1) ? wB1[s] : 0.f);
            acc = __builtin_amdgcn_wmma_f32_16x16x4_f32(
                      false, af, false, bf[s], (short)0, acc, false, false);
        }
    }

    // Positional-encoding bias: pe_term[d] = sum_k w[k]*pe[k][d], d = c mod 128
    const int d = c & (DDIM - 1);
    float pt = 0.f;
#pragma unroll
    for (int k = 0; k < KSZ; ++k) pt += w[k] * pe[k * DDIM + d];

    // Store: VGPR r holds output row (mt*16 + r + half*8), column c
#pragma unroll
    for (int r = 0; r < 8; ++r) {
        const int m = mt * 16 + r + half * 8;
        if (m < MOUT) {
            y[((long)b * MOUT + m) * CCH + c] = acc[r] + pt;
        }
    }
}

extern "C" void kernel_launch(void* const* d_in, const int* in_sizes, int n_in,
                              void* d_out, int out_size, void* d_ws, size_t ws_size,
                              hipStream_t stream) {
    (void)in_sizes; (void)n_in; (void)out_size; (void)d_ws; (void)ws_size;
    const float* x  = (const float*)d_in[0];
    const float* w  = (const float*)d_in[1];
    const float* pe = (const float*)d_in[2];
    float* y        = (float*)d_out;

    dim3 grid(32, 32, BATCH);   // (c-tile groups of 4 waves, m-tiles, batch)
    dim3 block(128, 1, 1);      // 4 waves, one 16x16 output tile per wave
    compresskv_wmma_kernel<<<grid, block, 0, stream>>>(x, w, pe, y);
}